// HL_LH_block_1_20624432956297
// MI455X (gfx1250) — compile-verified
//
#include <hip/hip_runtime.h>
#include <math.h>

#define NN   50000   // nodes
#define NE   800000  // edges
#define FI   128     // input features
#define HDIM 128     // heads*dim

typedef float v2f __attribute__((ext_vector_type(2)));
typedef float v8f __attribute__((ext_vector_type(8)));

__device__ __forceinline__ float lrelu(float x) { return x > 0.0f ? x : 0.01f * x; }
// order-preserving float<->int mapping (involution) for signed-int atomicMax on floats
__device__ __forceinline__ int   f2ord(float f) { int i = __float_as_int(f); return i >= 0 ? i : (i ^ 0x7fffffff); }
__device__ __forceinline__ float ord2f(int o)   { return __int_as_float(o >= 0 ? o : (o ^ 0x7fffffff)); }

// ---------------- init workspace ----------------
__global__ void k_init(float* agg, float* deg, float* denom, int* emax, float* hacc) {
  int idx = blockIdx.x * blockDim.x + threadIdx.x;
  if (idx < NN * HDIM) { agg[idx] = 0.0f; hacc[idx] = 0.0f; }
  if (idx < NN) deg[idx] = 0.0f;
  if (idx < NN * 4) { denom[idx] = 0.0f; emax[idx] = f2ord(-INFINITY); }
}

// ---------------- node GEMMs: f_ni = X*W_ni, f_nj = X*W_nj (y-dim selects W) ----------------
__global__ void k_node_gemm(const float* __restrict__ nfeats,
                            const float* __restrict__ Wni, const float* __restrict__ Wnj,
                            float* __restrict__ fni, float* __restrict__ fnj) {
  const int lane  = threadIdx.x & 31;
  const int group = blockIdx.x * (blockDim.x >> 5) + (threadIdx.x >> 5);
  if (group >= NN / 16) return;
  const float* W  = (blockIdx.y == 0) ? Wni : Wnj;
  float*       op = (blockIdx.y == 0) ? fni : fnj;
  const int hi = lane >> 4, lo = lane & 15;
  const int row0 = group * 16;
  v8f acc[8] = {};
  const float* arow = nfeats + (size_t)(row0 + lo) * FI + 2 * hi;
  for (int k = 0; k < FI; k += 4) {
    float2 av = *(const float2*)(arow + k);
    v2f a; a.x = av.x; a.y = av.y;
    const float* wp = W + (size_t)(k + 2 * hi) * HDIM + lo;
#pragma unroll
    for (int t = 0; t < 8; ++t) {
      v2f b; b.x = wp[t * 16]; b.y = wp[t * 16 + HDIM];
      acc[t] = __builtin_amdgcn_wmma_f32_16x16x4_f32(false, a, false, b, (short)0, acc[t], false, false);
    }
  }
#pragma unroll
  for (int r = 0; r < 8; ++r) {
    float* orow = op + (size_t)(row0 + r + 8 * hi) * HDIM + lo;
#pragma unroll
    for (int t = 0; t < 8; ++t) orow[t * 16] = acc[t][r];
  }
}

// ---------------- scatter efeats into agg[dst], count degree ----------------
__global__ void k_agg(const float* __restrict__ efeats, const int* __restrict__ dst,
                      float* __restrict__ agg, float* __restrict__ deg) {
  int idx = blockIdx.x * blockDim.x + threadIdx.x;
  if (idx >= NE * 32) return;
  int e = idx >> 5, q = idx & 31;
  int d = dst[e];
  float4 v = *(const float4*)(efeats + (size_t)e * FI + q * 4);
  float* ap = agg + (size_t)d * FI + q * 4;
  atomicAdd(ap + 0, v.x); atomicAdd(ap + 1, v.y);
  atomicAdd(ap + 2, v.z); atomicAdd(ap + 3, v.w);
  if (q == 0) atomicAdd(deg + d, 1.0f);
}

// ---------------- fused edge kernel: WMMA f_fij + gather + bias + lrelu + head-sum + logits + seg-max ----------------
__global__ void k_edge(const float* __restrict__ efeats, const float* __restrict__ Wfij,
                       const float* __restrict__ fni, const float* __restrict__ fnj,
                       const float* __restrict__ attn, const float* __restrict__ bias_e,
                       const int* __restrict__ src, const int* __restrict__ dst,
                       float* __restrict__ e_ws, int* __restrict__ emax,
                       float* __restrict__ fsum) {
  const int lane  = threadIdx.x & 31;
  const int group = blockIdx.x * (blockDim.x >> 5) + (threadIdx.x >> 5);
  if (group >= NE / 16) return;
  const int hi = lane >> 4, lo = lane & 15;
  const int eb = group * 16;
  v8f acc[8] = {};
  const float* arow = efeats + (size_t)(eb + lo) * FI + 2 * hi;
  for (int k = 0; k < FI; k += 4) {
    float2 av = *(const float2*)(arow + k);
    v2f a; a.x = av.x; a.y = av.y;
    const float* wp = Wfij + (size_t)(k + 2 * hi) * HDIM + lo;
#pragma unroll
    for (int t = 0; t < 8; ++t) {
      v2f b; b.x = wp[t * 16]; b.y = wp[t * 16 + HDIM];
      acc[t] = __builtin_amdgcn_wmma_f32_16x16x4_f32(false, a, false, b, (short)0, acc[t], false, false);
    }
  }
  // epilogue: f_out = lrelu(f_fij + f_ni[src] + f_nj[dst] + bias)
  int dsts[8];
#pragma unroll
  for (int r = 0; r < 8; ++r) {
    const int e = eb + r + 8 * hi;
    const int s = src[e];
    const int d = dst[e];
    dsts[r] = d;
    const float* pni = fni + (size_t)s * HDIM + lo;
    const float* pnj = fnj + (size_t)d * HDIM + lo;
#pragma unroll
    for (int t = 0; t < 8; ++t)
      acc[t][r] = lrelu(acc[t][r] + pni[t * 16] + pnj[t * 16] + bias_e[t * 16 + lo]);
  }
  // f_out.sum(axis=heads) -> second output [E,32]
#pragma unroll
  for (int r = 0; r < 8; ++r) {
    const int e = eb + r + 8 * hi;
    fsum[(size_t)e * 32 + lo]      = acc[0][r] + acc[2][r] + acc[4][r] + acc[6][r];
    fsum[(size_t)e * 32 + 16 + lo] = acc[1][r] + acc[3][r] + acc[5][r] + acc[7][r];
  }
  // per-head attention logits (reduce 16 lanes per half-wave) + segment max via ordered-int atomicMax
#pragma unroll
  for (int r = 0; r < 8; ++r) {
#pragma unroll
    for (int h = 0; h < 4; ++h) {
      float p = acc[2 * h][r] * attn[h * 32 + lo] + acc[2 * h + 1][r] * attn[h * 32 + 16 + lo];
      p += __shfl_xor(p, 1, 16);
      p += __shfl_xor(p, 2, 16);
      p += __shfl_xor(p, 4, 16);
      p += __shfl_xor(p, 8, 16);
      if (lo == 0) {
        const int e = eb + r + 8 * hi;
        e_ws[(size_t)e * 4 + h] = p;
        atomicMax(&emax[dsts[r] * 4 + h], f2ord(p));
      }
    }
  }
}

// ---------------- softmax numerator + denominator ----------------
__global__ void k_exp(const int* __restrict__ dst, const int* __restrict__ emax,
                      float* __restrict__ e_ws, float* __restrict__ denom) {
  int idx = blockIdx.x * blockDim.x + threadIdx.x;
  if (idx >= NE * 4) return;
  int e = idx >> 2, h = idx & 3;
  int d = dst[e];
  float ex = __expf(e_ws[idx] - ord2f(emax[d * 4 + h]));
  e_ws[idx] = ex;
  atomicAdd(&denom[d * 4 + h], ex);
}

// ---------------- agg /= max(deg,1) ----------------
__global__ void k_aggnorm(float* __restrict__ agg, const float* __restrict__ deg) {
  int idx = blockIdx.x * blockDim.x + threadIdx.x;
  if (idx >= NN * HDIM) return;
  agg[idx] = agg[idx] / fmaxf(deg[idx >> 7], 1.0f);
}

// ---------------- node transform: h = [nfeats || agg] @ W_node + b (K=256) ----------------
__global__ void k_hgemm(const float* __restrict__ nfeats, const float* __restrict__ agg,
                        const float* __restrict__ Wn, const float* __restrict__ bn,
                        float* __restrict__ hmat) {
  const int lane  = threadIdx.x & 31;
  const int group = blockIdx.x * (blockDim.x >> 5) + (threadIdx.x >> 5);
  if (group >= NN / 16) return;
  const int hi = lane >> 4, lo = lane & 15;
  const int row0 = group * 16;
  v8f acc[8] = {};
  const float* a0 = nfeats + (size_t)(row0 + lo) * FI + 2 * hi;
  const float* a1 = agg    + (size_t)(row0 + lo) * FI + 2 * hi;
  for (int k = 0; k < 2 * FI; k += 4) {
    const float* ap = (k < FI) ? (a0 + k) : (a1 + (k - FI));
    float2 av = *(const float2*)ap;
    v2f a; a.x = av.x; a.y = av.y;
    const float* wp = Wn + (size_t)(k + 2 * hi) * HDIM + lo;
#pragma unroll
    for (int t = 0; t < 8; ++t) {
      v2f b; b.x = wp[t * 16]; b.y = wp[t * 16 + HDIM];
      acc[t] = __builtin_amdgcn_wmma_f32_16x16x4_f32(false, a, false, b, (short)0, acc[t], false, false);
    }
  }
#pragma unroll
  for (int r = 0; r < 8; ++r) {
    float* orow = hmat + (size_t)(row0 + r + 8 * hi) * HDIM + lo;
#pragma unroll
    for (int t = 0; t < 8; ++t) orow[t * 16] = acc[t][r] + bn[t * 16 + lo];
  }
}

// ---------------- attention-weighted scatter: hacc[dst] += a * h[src] ----------------
__global__ void k_msg(const int* __restrict__ src, const int* __restrict__ dst,
                      const float* __restrict__ hmat, const float* __restrict__ e_ws,
                      const float* __restrict__ denom, float* __restrict__ hacc) {
  int idx = blockIdx.x * blockDim.x + threadIdx.x;
  if (idx >= NE * 128) return;
  int e = idx >> 7, c = idx & 127;
  int h = c >> 5;
  int d = dst[e];
  float a = e_ws[(size_t)e * 4 + h] / denom[d * 4 + h];
  atomicAdd(&hacc[(size_t)d * HDIM + c], hmat[(size_t)src[e] * HDIM + c] * a);
}

// ---------------- h_out = sum_h lrelu(hacc) ----------------
__global__ void k_out(const float* __restrict__ hacc, float* __restrict__ out) {
  int idx = blockIdx.x * blockDim.x + threadIdx.x;
  if (idx >= NN * 32) return;
  int n = idx >> 5, d = idx & 31;
  const float* p = hacc + (size_t)n * HDIM + d;
  out[idx] = lrelu(p[0]) + lrelu(p[32]) + lrelu(p[64]) + lrelu(p[96]);
}

extern "C" void kernel_launch(void* const* d_in, const int* in_sizes, int n_in,
                              void* d_out, int out_size, void* d_ws, size_t ws_size,
                              hipStream_t stream) {
  (void)in_sizes; (void)n_in; (void)out_size; (void)ws_size;
  const float* nfeats = (const float*)d_in[0];
  const float* efeats = (const float*)d_in[1];
  const int*   src    = (const int*)d_in[2];
  const int*   dst    = (const int*)d_in[3];
  const float* W_ni   = (const float*)d_in[4];
  const float* W_nj   = (const float*)d_in[5];
  const float* W_fij  = (const float*)d_in[6];
  const float* attn   = (const float*)d_in[7];
  const float* bias_e = (const float*)d_in[8];
  const float* W_node = (const float*)d_in[9];
  const float* b_node = (const float*)d_in[10];
  float* out = (float*)d_out;
  float* ws  = (float*)d_ws;

  // workspace layout (floats): ~142.6 MB total
  float* fni   = ws;
  float* fnj   = fni  + (size_t)NN * HDIM;
  float* agg   = fnj  + (size_t)NN * HDIM;
  float* hmat  = agg  + (size_t)NN * HDIM;
  float* hacc  = hmat + (size_t)NN * HDIM;
  float* deg   = hacc + (size_t)NN * HDIM;
  float* denom = deg  + NN;
  float* e_ws  = denom + (size_t)NN * 4;
  int*   emax  = (int*)(e_ws + (size_t)NE * 4);

  float* fsum = out + (size_t)NN * 32;  // second output region [E,32]

  const int T = 256;
  k_init   <<<(NN * HDIM + T - 1) / T, T, 0, stream>>>(agg, deg, denom, emax, hacc);
  {
    dim3 grid((NN / 16 + 3) / 4, 2);
    k_node_gemm<<<grid, 128, 0, stream>>>(nfeats, W_ni, W_nj, fni, fnj);
  }
  k_agg    <<<(NE * 32 + T - 1) / T, T, 0, stream>>>(efeats, dst, agg, deg);
  k_edge   <<<(NE / 16 + 3) / 4, 128, 0, stream>>>(efeats, W_fij, fni, fnj, attn, bias_e,
                                                   src, dst, e_ws, emax, fsum);
  k_exp    <<<(NE * 4 + T - 1) / T, T, 0, stream>>>(dst, emax, e_ws, denom);
  k_aggnorm<<<(NN * HDIM + T - 1) / T, T, 0, stream>>>(agg, deg);
  k_hgemm  <<<(NN / 16 + 3) / 4, 128, 0, stream>>>(nfeats, agg, W_node, b_node, hmat);
  k_msg    <<<(NE * 128 + T - 1) / T, T, 0, stream>>>(src, dst, hmat, e_ws, denom, hacc);
  k_out    <<<(NN * 32 + T - 1) / T, T, 0, stream>>>(hacc, out);
}